// SpectrogramAugmentation_4440996184687
// MI455X (gfx1250) — compile-verified
//
#include <hip/hip_runtime.h>

typedef float v4f __attribute__((ext_vector_type(4)));

// Problem constants (match reference)
constexpr int   Bc = 64, Cc = 2, Fc = 128, Tc = 2048;
constexpr int   CFT = Cc * Fc * Tc;                 // 524288 floats per batch
constexpr float FREQ_MASK_PARAM = 20.0f;
constexpr float TIME_MASK_PARAM = 40.0f;
constexpr float NOISE_STD = 0.01f;
constexpr float P_APPLY  = 0.5f;

// Tiling: 256 threads * 4 iters * float4 = 4096 floats per block
constexpr int THREADS          = 256;
constexpr int ITERS            = 4;
constexpr int FLOATS_PER_ITER  = THREADS * 4;       // 1024
constexpr int FLOATS_PER_BLOCK = FLOATS_PER_ITER * ITERS; // 4096
constexpr int BLOCKS_PER_BATCH = CFT / FLOATS_PER_BLOCK;  // 128

__global__ __launch_bounds__(THREADS)
void specaug_kernel(const float* __restrict__ spec,
                    const float* __restrict__ apply_u,
                    const float* __restrict__ f_width_u,
                    const float* __restrict__ f_start_u,
                    const float* __restrict__ t_width_u,
                    const float* __restrict__ t_start_u,
                    const float* __restrict__ noise,
                    float*       __restrict__ out)
{
    const int b   = blockIdx.y;                 // batch (uniform per block)
    const int tid = threadIdx.x;
    const long base = (long)b * CFT + (long)blockIdx.x * FLOATS_PER_BLOCK;

    const float* sp = spec  + base;
    const float* np = noise + base;
    float*       op = out   + base;

    const bool apply = (apply_u[b] <= P_APPLY); // uniform branch per block

    // Streaming lookahead: warm GL2 with the NEXT block's tile (16 KB/tensor).
    // 256 lanes * 64B = 16 KB = FLOATS_PER_BLOCK * 4B. Emits global_prefetch_b8.
    if (blockIdx.x + 1 < BLOCKS_PER_BATCH) {
        const char* ps = (const char*)(sp + FLOATS_PER_BLOCK) + tid * 64;
        __builtin_prefetch(ps, 0, 1);
        if (apply) {
            const char* pn = (const char*)(np + FLOATS_PER_BLOCK) + tid * 64;
            __builtin_prefetch(pn, 0, 1);
        }
    }

    if (!apply) {
        // Pure NT streaming copy: out = spec (noise never touched).
#pragma unroll
        for (int j = 0; j < ITERS; ++j) {
            const v4f* p = (const v4f*)(sp + j * FLOATS_PER_ITER) + tid;
            v4f v = __builtin_nontemporal_load(p);
            __builtin_nontemporal_store(v, (v4f*)(op + j * FLOATS_PER_ITER) + tid);
        }
        return;
    }

    // ---- Per-batch mask parameters (uniform; scalar-unit math) ----
    // freq masks: w = floor(u*20); s0 = floor(u2 * (F - w))          (no clamp)
    // time masks: w = floor(u*40); s0 = floor(u2 * max(T - w, 1))    (clamp >= 1)
    const int fw0 = (int)floorf(f_width_u[b * 2 + 0] * FREQ_MASK_PARAM);
    const int fw1 = (int)floorf(f_width_u[b * 2 + 1] * FREQ_MASK_PARAM);
    const int fs0 = (int)floorf(f_start_u[b * 2 + 0] * (float)(Fc - fw0));
    const int fs1 = (int)floorf(f_start_u[b * 2 + 1] * (float)(Fc - fw1));

    const int tw0 = (int)floorf(t_width_u[b * 2 + 0] * TIME_MASK_PARAM);
    const int tw1 = (int)floorf(t_width_u[b * 2 + 1] * TIME_MASK_PARAM);
    const int ts0 = (int)floorf(t_start_u[b * 2 + 0] * fmaxf((float)(Tc - tw0), 1.0f));
    const int ts1 = (int)floorf(t_start_u[b * 2 + 1] * fmaxf((float)(Tc - tw1), 1.0f));

    const int fe0 = fs0 + fw0, fe1 = fs1 + fw1;
    const int te0 = ts0 + tw0, te1 = ts1 + tw1;

    // ---- Batch all NT b128 loads first (8 outstanding on LOADcnt) ----
    v4f s[ITERS], n[ITERS];
#pragma unroll
    for (int j = 0; j < ITERS; ++j) {
        s[j] = __builtin_nontemporal_load((const v4f*)(sp + j * FLOATS_PER_ITER) + tid);
        n[j] = __builtin_nontemporal_load((const v4f*)(np + j * FLOATS_PER_ITER) + tid);
    }

    // ---- Mask + noise, NT store ----
#pragma unroll
    for (int j = 0; j < ITERS; ++j) {
        const int pos = blockIdx.x * FLOATS_PER_BLOCK + j * FLOATS_PER_ITER + tid * 4;
        const int rem = pos & (Fc * Tc - 1);    // within one channel plane
        const int f   = rem >> 11;              // T = 2048 = 2^11
        const int t0  = rem & (Tc - 1);         // 4-aligned; same f for all 4 lanes' elems

        const bool fm = (f >= fs0 && f < fe0) || (f >= fs1 && f < fe1);

        v4f r;
#pragma unroll
        for (int k = 0; k < 4; ++k) {
            const int t = t0 + k;
            const bool tm = (t >= ts0 && t < te0) || (t >= ts1 && t < te1);
            const float v = (fm || tm) ? 0.0f : s[j][k];
            r[k] = __builtin_fmaf(n[j][k], NOISE_STD, v);
        }
        __builtin_nontemporal_store(r, (v4f*)(op + j * FLOATS_PER_ITER) + tid);
    }
}

extern "C" void kernel_launch(void* const* d_in, const int* in_sizes, int n_in,
                              void* d_out, int out_size, void* d_ws, size_t ws_size,
                              hipStream_t stream) {
    (void)in_sizes; (void)n_in; (void)out_size; (void)d_ws; (void)ws_size;
    const float* spec      = (const float*)d_in[0];
    const float* apply_u   = (const float*)d_in[1];
    const float* f_width_u = (const float*)d_in[2];
    const float* f_start_u = (const float*)d_in[3];
    const float* t_width_u = (const float*)d_in[4];
    const float* t_start_u = (const float*)d_in[5];
    const float* noise     = (const float*)d_in[6];
    float*       out       = (float*)d_out;

    dim3 grid(BLOCKS_PER_BATCH, Bc, 1);   // (128, 64) blocks, 8 waves each
    specaug_kernel<<<grid, THREADS, 0, stream>>>(
        spec, apply_u, f_width_u, f_start_u, t_width_u, t_start_u, noise, out);
}